// MixtureOfExperts_73521250173677
// MI455X (gfx1250) — compile-verified
//
#include <hip/hip_runtime.h>
#include <math.h>

#define T_TOK   16384
#define H_DIM   768
#define I_DIM   1536
#define N_EXP   8
#define TILE_M  32            // 2 x 16-row WMMA sub-tiles
#define CI      256           // I-chunk per iteration (16 waves x 16 cols)
#define XS_STRIDE 772         // 768 + 4 (bank skew: 772 % 64 == 4)
#define HS_STRIDE 260         // 256 + 4
#define MAX_TILES 512         // ceil(16384 / 32): per-expert worst case

typedef float v2f __attribute__((ext_vector_type(2)));
typedef float v8f __attribute__((ext_vector_type(8)));

__device__ inline v8f wmma4(v2f a, v2f b, v8f c) {
    // V_WMMA_F32_16X16X4_F32 : 16x16 fp32 C/D, K=4 fp32 A/B
    return __builtin_amdgcn_wmma_f32_16x16x4_f32(
        /*neg_a=*/false, a, /*neg_b=*/false, b,
        /*c_mod=*/(short)0, c, /*reuse_a=*/false, /*reuse_b=*/false);
}

__device__ inline float silu_fast(float g) {
    // g * rcp(1+exp(-g)) : v_exp_f32 + v_add + v_rcp_f32 + v_mul (no IEEE div chain)
    return g * __builtin_amdgcn_rcpf(1.0f + __expf(-g));
}

// ---------------------------------------------------------------------------
// Router: one wave32 per token. logits = x[t] . router_w ; softmax; top-2;
// bucket append + loss statistics.
// ---------------------------------------------------------------------------
__global__ __launch_bounds__(256) void moe_router(
    const float* __restrict__ x, const float* __restrict__ rw,
    int* __restrict__ counts, float* __restrict__ sum_prob,
    float* __restrict__ sum_lse2,
    int* __restrict__ tok_list, float* __restrict__ w_list)
{
    const int lane = threadIdx.x & 31;
    const int wave = threadIdx.x >> 5;
    const int tok  = blockIdx.x * 8 + wave;
    if (tok >= T_TOK) return;

    float acc[N_EXP];
    #pragma unroll
    for (int e = 0; e < N_EXP; ++e) acc[e] = 0.0f;

    const float* xr = x + (size_t)tok * H_DIM;
    for (int h = lane; h < H_DIM; h += 32) {            // coalesced b32 loads
        float xv = xr[h];
        const float4* wr = (const float4*)(rw + (size_t)h * N_EXP);
        float4 w0 = wr[0], w1 = wr[1];                   // 24KB table, cache-hot
        acc[0] += xv * w0.x; acc[1] += xv * w0.y;
        acc[2] += xv * w0.z; acc[3] += xv * w0.w;
        acc[4] += xv * w1.x; acc[5] += xv * w1.y;
        acc[6] += xv * w1.z; acc[7] += xv * w1.w;
    }
    #pragma unroll
    for (int e = 0; e < N_EXP; ++e) {
        float v = acc[e];
        v += __shfl_xor(v, 16, 32);
        v += __shfl_xor(v,  8, 32);
        v += __shfl_xor(v,  4, 32);
        v += __shfl_xor(v,  2, 32);
        v += __shfl_xor(v,  1, 32);
        acc[e] = v;
    }
    if (lane == 0) {
        float mx = acc[0];
        #pragma unroll
        for (int e = 1; e < N_EXP; ++e) mx = fmaxf(mx, acc[e]);
        float p[N_EXP];
        float s = 0.0f;
        #pragma unroll
        for (int e = 0; e < N_EXP; ++e) { p[e] = __expf(acc[e] - mx); s += p[e]; }
        float inv = 1.0f / s;
        float lse = mx + __logf(s);
        #pragma unroll
        for (int e = 0; e < N_EXP; ++e) p[e] *= inv;

        int i0 = 0; float p0 = p[0];
        #pragma unroll
        for (int e = 1; e < N_EXP; ++e) if (p[e] > p0) { p0 = p[e]; i0 = e; }
        int i1 = (i0 == 0) ? 1 : 0; float p1 = p[i1];
        #pragma unroll
        for (int e = 0; e < N_EXP; ++e)
            if (e != i0 && p[e] > p1) { p1 = p[e]; i1 = e; }

        float wsum = p0 + p1;
        float rw0 = p0 / wsum, rw1 = p1 / wsum;

        atomicAdd(sum_lse2, lse * lse);
        #pragma unroll
        for (int e = 0; e < N_EXP; ++e) atomicAdd(&sum_prob[e], p[e]);

        int pos0 = atomicAdd(&counts[i0], 1);
        tok_list[i0 * T_TOK + pos0] = tok;
        w_list [i0 * T_TOK + pos0] = rw0;
        int pos1 = atomicAdd(&counts[i1], 1);
        tok_list[i1 * T_TOK + pos1] = tok;
        w_list [i1 * T_TOK + pos1] = rw1;
    }
}

// ---------------------------------------------------------------------------
// Loss finalize: lb = coef*E*sum_e f_e*P_e ; z = coef*mean(lse^2)
// ---------------------------------------------------------------------------
__global__ void moe_losses(const int* __restrict__ counts,
                           const float* __restrict__ sum_prob,
                           const float* __restrict__ sum_lse2,
                           float* __restrict__ out)
{
    if (threadIdx.x == 0 && blockIdx.x == 0) {
        float lb = 0.0f;
        for (int e = 0; e < N_EXP; ++e) {
            float f = (float)counts[e] / (float)(T_TOK * 2);
            float P = sum_prob[e] / (float)T_TOK;
            lb += f * P;
        }
        out[(size_t)T_TOK * H_DIM]     = 1e-3f * (float)N_EXP * lb;
        out[(size_t)T_TOK * H_DIM + 1] = 1e-3f * sum_lse2[0] / (float)T_TOK;
    }
}

// ---------------------------------------------------------------------------
// Grouped expert MLP: block = (expert, 32-token tile), 512 threads = 16 waves.
// Each wave: gate+up for one 16-col subtile x BOTH 16-row M-subtiles (B reuse),
// then down-proj for 3 N-tiles x both M-subtiles (B reuse across M).
// All matmuls on V_WMMA_F32_16X16X4_F32. Weighted atomic scatter to out.
// ---------------------------------------------------------------------------
__global__ __launch_bounds__(512, 1) void moe_expert_mlp(
    const float* __restrict__ x,
    const float* __restrict__ wg, const float* __restrict__ wu,
    const float* __restrict__ wd,
    const int* __restrict__ counts, const int* __restrict__ tok_list,
    const float* __restrict__ w_list, float* __restrict__ out)
{
    __shared__ float xs[TILE_M * XS_STRIDE];   // 32 x 768 token tile (~99KB)
    __shared__ float hs[TILE_M * HS_STRIDE];   // 32 x 256 silu(g)*u chunk (~33KB)
    __shared__ int   toks[TILE_M];
    __shared__ float wts [TILE_M];

    const int e    = blockIdx.x >> 9;            // expert
    const int tile = blockIdx.x & (MAX_TILES - 1);
    const int cnt  = counts[e];
    const int row0 = tile * TILE_M;
    if (row0 >= cnt) return;

    const int tid  = threadIdx.x;
    const int wave = tid >> 5;                   // 0..15
    const int lane = tid & 31;
    const int lm   = lane & 15;                  // A-row / B-col within 16-tile
    const int koff = (lane >> 4) << 1;           // 0 or 2: K sub-offset

    if (tid < TILE_M) {
        int r = row0 + tid;
        toks[tid] = (r < cnt) ? tok_list[e * T_TOK + r] : 0;
        wts [tid] = (r < cnt) ? w_list [e * T_TOK + r] : 0.0f;   // pad rows: weight 0
    }
    __syncthreads();

    // stage 32 token rows of x into LDS (float4, coalesced); 2 rows per wave
    for (int r = wave; r < TILE_M; r += 16) {
        const float4* src = (const float4*)(x + (size_t)toks[r] * H_DIM);
        float4* dst = (float4*)&xs[r * XS_STRIDE];
        for (int c = lane; c < H_DIM / 4; c += 32) dst[c] = src[c];
    }

    v8f zero = {};
    v8f dacc[2][3];                              // [m-subtile][n-tile], 48 cols/wave
    #pragma unroll
    for (int mt = 0; mt < 2; ++mt)
        #pragma unroll
        for (int j = 0; j < 3; ++j) dacc[mt][j] = zero;

    const size_t ebase = (size_t)e * H_DIM * I_DIM;

    for (int i0 = 0; i0 < I_DIM; i0 += CI) {
        __syncthreads();   // xs ready (iter 0) / previous hs fully consumed

        // ---- gate & up for cols [i0 + wave*16, +16), both M-subtiles ----
        const int col0 = i0 + (wave << 4);
        v8f ag0 = zero, au0 = zero, ag1 = zero, au1 = zero;
        const float* gp   = wg + ebase + (size_t)koff * I_DIM + col0 + lm;
        const float* up   = wu + ebase + (size_t)koff * I_DIM + col0 + lm;
        const float* xsr0 = &xs[lm * XS_STRIDE + koff];
        const float* xsr1 = xsr0 + 16 * XS_STRIDE;
        #pragma unroll 2
        for (int k = 0; k < H_DIM; k += 4) {     // 192 iters, 4 WMMA each
            v2f a0 = *(const v2f*)(xsr0 + k);
            v2f a1 = *(const v2f*)(xsr1 + k);
            v2f bg; bg.x = gp[0]; bg.y = gp[I_DIM];
            v2f bu; bu.x = up[0]; bu.y = up[I_DIM];
            ag0 = wmma4(a0, bg, ag0);
            ag1 = wmma4(a1, bg, ag1);            // B reused across M-subtiles
            au0 = wmma4(a0, bu, au0);
            au1 = wmma4(a1, bu, au1);
            gp += (size_t)4 * I_DIM;
            up += (size_t)4 * I_DIM;
        }
        // silu(g) * u -> hs  (C-layout: vgpr v, lane -> row v+(lane/16)*8, col lane%16)
        #pragma unroll
        for (int v = 0; v < 8; ++v) {
            int m = v + ((lane >> 4) << 3);
            int c = (wave << 4) + lm;
            hs[m * HS_STRIDE + c]        = silu_fast(ag0[v]) * au0[v];
            hs[(m + 16) * HS_STRIDE + c] = silu_fast(ag1[v]) * au1[v];
        }
        __syncthreads();

        // ---- down proj: out[32 x 768] += h_chunk . Wd ; 3 n-tiles x 2 m ----
        const float* hsr0 = &hs[lm * HS_STRIDE + koff];
        const float* hsr1 = hsr0 + 16 * HS_STRIDE;
        const float* wdp  = wd + ((size_t)e * I_DIM + i0 + koff) * H_DIM
                               + wave * 48 + lm;
        #pragma unroll 2
        for (int k = 0; k < CI; k += 4) {        // 64 iters, 6 WMMA each
            v2f a0 = *(const v2f*)(hsr0 + k);
            v2f a1 = *(const v2f*)(hsr1 + k);
            #pragma unroll
            for (int j = 0; j < 3; ++j) {
                v2f b; b.x = wdp[j * 16]; b.y = wdp[H_DIM + j * 16];
                dacc[0][j] = wmma4(a0, b, dacc[0][j]);
                dacc[1][j] = wmma4(a1, b, dacc[1][j]);   // B reused across M
            }
            wdp += (size_t)4 * H_DIM;
        }
    }

    // weighted scatter (top-2 experts collide on tokens -> atomic f32 add)
    #pragma unroll
    for (int mt = 0; mt < 2; ++mt) {
        #pragma unroll
        for (int j = 0; j < 3; ++j) {
            int col = wave * 48 + j * 16 + lm;
            #pragma unroll
            for (int v = 0; v < 8; ++v) {
                int m = mt * 16 + v + ((lane >> 4) << 3);
                atomicAdd(&out[(size_t)toks[m] * H_DIM + col],
                          wts[m] * dacc[mt][j][v]);
            }
        }
    }
}

// ---------------------------------------------------------------------------
extern "C" void kernel_launch(void* const* d_in, const int* in_sizes, int n_in,
                              void* d_out, int out_size, void* d_ws, size_t ws_size,
                              hipStream_t stream) {
    (void)in_sizes; (void)n_in; (void)ws_size;
    const float* x  = (const float*)d_in[0];   // [T,768]
    const float* rw = (const float*)d_in[1];   // [768,8]
    const float* wg = (const float*)d_in[2];   // [8,768,1536]
    const float* wu = (const float*)d_in[3];   // [8,768,1536]
    const float* wd = (const float*)d_in[4];   // [8,1536,768]
    float* out = (float*)d_out;                // [T*768] ++ [lb, z]

    char* ws = (char*)d_ws;
    int*   counts   = (int*)(ws + 0);                                   // 8 ints
    float* sum_prob = (float*)(ws + 64);                                // 8 f32
    float* sum_lse2 = (float*)(ws + 128);                               // 1 f32
    int*   tok_list = (int*)(ws + 256);                                 // 8*16384 int
    float* w_list   = (float*)(ws + 256 + sizeof(int) * N_EXP * T_TOK); // 8*16384 f32

    hipMemsetAsync(d_ws, 0, 256, stream);                         // counters/stats
    hipMemsetAsync(d_out, 0, (size_t)out_size * sizeof(float), stream);

    moe_router    <<<T_TOK / 8,          256, 0, stream>>>(x, rw, counts, sum_prob,
                                                           sum_lse2, tok_list, w_list);
    moe_losses    <<<1,                  1,   0, stream>>>(counts, sum_prob, sum_lse2, out);
    moe_expert_mlp<<<N_EXP * MAX_TILES,  512, 0, stream>>>(x, wg, wu, wd, counts,
                                                           tok_list, w_list, out);
}